// SpatialEncoding_73804718015010
// MI455X (gfx1250) — compile-verified
//
#include <hip/hip_runtime.h>
#include <hip/hip_bf16.h>

// ---------------------------------------------------------------------------
// SpatialEncoding scatter for MI455X (gfx1250).
//   out (N x N fp32) = 0;  out[src[i], dst[i]] = (len[i] > 0) ? b[clamp(min(len[i],B)-1, 0, B-1)] : 0
// Memory-bound: ~256MB zero-fill + ~48MB index reads + 4M scattered 4B writes.
// No matrix math -> no WMMA; use gfx1250 async store-from-LDS for the fill.
// ---------------------------------------------------------------------------

#define HAS_ASYNC_ST  __has_builtin(__builtin_amdgcn_global_store_async_from_lds_b128)
#define HAS_WAIT_ASY  __has_builtin(__builtin_amdgcn_s_wait_asynccnt)

typedef int v4i __attribute__((vector_size(16)));
typedef __attribute__((address_space(3))) v4i lds_v4i;

// Zero-fill out[0 .. total) using 16B stores. Primary path: async store from a
// 16B LDS line of zeros (gfx1250 GLOBAL_STORE_ASYNC_FROM_LDS_B128, ASYNCcnt).
__global__ void se_zero_kernel(float* __restrict__ out, long long n4, long long total)
{
    long long tid    = (long long)blockIdx.x * blockDim.x + threadIdx.x;
    long long stride = (long long)gridDim.x * blockDim.x;

#if HAS_ASYNC_ST
    __shared__ float zline[4];
    if (threadIdx.x < 4) zline[threadIdx.x] = 0.0f;
    __syncthreads();
    lds_v4i* lsrc = (lds_v4i*)&zline[0];
    for (long long i = tid; i < n4; i += stride) {
        __builtin_amdgcn_global_store_async_from_lds_b128((v4i*)(out + 4 * i), lsrc, 0, 0);
    }
#if HAS_WAIT_ASY
    __builtin_amdgcn_s_wait_asynccnt(0);
#endif
#else
    const float4 z = make_float4(0.f, 0.f, 0.f, 0.f);
    float4* __restrict__ o4 = (float4*)out;
    for (long long i = tid; i < n4; i += stride) {
        o4[i] = z;
    }
#endif

    // Tail (total not divisible by 4): handled by the first thread.
    if (tid == 0) {
        for (long long i = n4 * 4; i < total; ++i) out[i] = 0.0f;
    }
}

// One thread per (src,dst,len) triple.
__global__ void se_scatter_kernel(const float* __restrict__ b, int bmax,
                                  const int* __restrict__ src,
                                  const int* __restrict__ dst,
                                  const int* __restrict__ len,
                                  float* __restrict__ out,
                                  int P, int N)
{
    int i = blockIdx.x * blockDim.x + threadIdx.x;
    if (i >= P) return;

    int L   = len[i];
    int idx = (L < bmax ? L : bmax) - 1;
    idx     = idx < 0 ? 0 : (idx > bmax - 1 ? bmax - 1 : idx);
    float v = (L > 0) ? b[idx] : 0.0f;

    long long o = (long long)src[i] * (long long)N + (long long)dst[i];
    out[o] = v;
}

extern "C" void kernel_launch(void* const* d_in, const int* in_sizes, int n_in,
                              void* d_out, int out_size, void* d_ws, size_t ws_size,
                              hipStream_t stream)
{
    // Inputs (setup_inputs order): x [N*D] (unused), b [MAX_PATH],
    // path_src [P], path_dst [P], path_len [P].
    const float* b    = (const float*)d_in[1];
    const int*   psrc = (const int*)d_in[2];
    const int*   pdst = (const int*)d_in[3];
    const int*   plen = (const int*)d_in[4];

    const int bmax = in_sizes[1];
    const int P    = in_sizes[2];

    // Recover N from out_size = N*N (integer sqrt, exact for this problem).
    long long N = 1;
    while (N * N < (long long)out_size) ++N;

    float* out = (float*)d_out;

    // ---- Pass 1: zero-fill 256MB with 16B stores (RT hint keeps lines in the
    //      192MB L2 so the scatter's RMWs mostly hit L2, not HBM). ----
    const int  threads = 256;
    long long  n4      = (long long)out_size / 4;
    long long  zblocks = (n4 + threads - 1) / threads;
    if (zblocks < 1) zblocks = 1;
    if (zblocks > 1048576) zblocks = 1048576;  // grid-stride loop covers the rest
    se_zero_kernel<<<(unsigned)zblocks, threads, 0, stream>>>(out, n4, (long long)out_size);

    // ---- Pass 2: scatter the 4M values. ----
    int sblocks = (P + threads - 1) / threads;
    se_scatter_kernel<<<sblocks, threads, 0, stream>>>(b, bmax, psrc, pdst, plen,
                                                       out, P, (int)N);
}